// GraphConvolution_76450417869342
// MI455X (gfx1250) — compile-verified
//
#include <hip/hip_runtime.h>

#define N_NODES 50000
#define N_EDGES 800000
#define F_IN    128
#define F_OUT   64
#define EDGES_PER_WAVE 128   // divides N_EDGES exactly (800000 = 6250 * 128)

typedef __attribute__((ext_vector_type(2))) float v2f;
typedef __attribute__((ext_vector_type(8))) float v8f;

// ---------------------------------------------------------------------------
// Phase 1: pre_sup = x @ W  via V_WMMA_F32_16X16X4_F32 (exact fp32 math).
// One wave32 per 16x16 output tile; K=128 consumed in 32 WMMA steps of 4.
// ---------------------------------------------------------------------------
__global__ __launch_bounds__(256) void gc_gemm_wmma(
    const float* __restrict__ x, const float* __restrict__ w,
    float* __restrict__ pre)
{
  const int gwave = (int)((blockIdx.x * blockDim.x + threadIdx.x) >> 5);
  const int lane  = (int)(threadIdx.x & 31);
  const int tiles = (N_NODES / 16) * (F_OUT / 16);   // 3125 * 4 = 12500
  if (gwave >= tiles) return;                        // wave-uniform: EXEC stays all-1s
  const int tm   = gwave >> 2;                       // node-tile
  const int tn   = gwave & 3;                        // feature-tile
  const int half = lane >> 4;                        // which 16-lane half
  const int idx  = lane & 15;

  // A (16x4, M=lane): VGPR0 holds K = k0+2*half, VGPR1 holds K = k0+2*half+1
  const float* __restrict__ xrow = x + (tm * 16 + idx) * F_IN;
  // B (4x16, N=lane): same K striping across lane halves / VGPRs
  const float* __restrict__ wcol = w + tn * 16 + idx;

  v8f c = {};
#pragma unroll
  for (int k0 = 0; k0 < F_IN; k0 += 4) {
    const int ka = k0 + 2 * half;
    v2f a, b;
    a.x = xrow[ka];
    a.y = xrow[ka + 1];
    b.x = wcol[ka * F_OUT];
    b.y = wcol[(ka + 1) * F_OUT];
    // 8 args: (neg_a, A, neg_b, B, c_mod, C, reuse_a, reuse_b)
    c = __builtin_amdgcn_wmma_f32_16x16x4_f32(false, a, false, b,
                                              (short)0, c, false, false);
  }

  // C/D layout: VGPR v holds row M = v + 8*half, column N = idx
  float* __restrict__ o = pre + (tm * 16 + 8 * half) * F_OUT + tn * 16 + idx;
#pragma unroll
  for (int v = 0; v < 8; ++v) o[v * F_OUT] = c[v];
}

// ---------------------------------------------------------------------------
// Phase 2: out[r] += val[e] * pre[col[e]]  (edge_row is sorted).
// One wave per 128-edge chunk. Per 32-edge batch, each lane cooperatively
// loads one edge's (row, col, val); per-edge values are broadcast with
// __shfl (lane-permute, no memory traffic). Lane L accumulates features
// 2L and 2L+1 so the pre_sup gather is a single coalesced b64 per lane
// (256 contiguous bytes per edge per wave, L2-resident on MI455X).
// Atomic flushes only at row changes (~16 edges/row average).
// ---------------------------------------------------------------------------
__global__ __launch_bounds__(256) void gc_aggregate(
    const float*  __restrict__ pre,
    const int*    __restrict__ erow,
    const int*    __restrict__ ecol,
    const float*  __restrict__ eval,
    float*        __restrict__ out)
{
  const int gwave = (int)((blockIdx.x * blockDim.x + threadIdx.x) >> 5);
  const int lane  = (int)(threadIdx.x & 31);
  const int start = gwave * EDGES_PER_WAVE;
  if (start >= N_EDGES) return;            // wave-uniform

  const float2* __restrict__ pre2 = (const float2*)pre;   // [N_NODES][32] of float2
  float accx = 0.f, accy = 0.f;
  int cur = erow[start];                   // wave-uniform (same addr all lanes)

  for (int base = start; base < start + EDGES_PER_WAVE; base += 32) {
    // cooperative metadata load: one edge per lane
    const int   r32 = erow[base + lane];
    const int   c32 = ecol[base + lane];
    const float v32 = eval[base + lane];

    for (int j = 0; j < 32; ++j) {         // uniform shuffle index
      const int   r = __shfl(r32, j, 32);
      const int   cN = __shfl(c32, j, 32);
      const float v = __shfl(v32, j, 32);
      if (r != cur) {                      // wave-uniform branch
        atomicAdd(&out[cur * F_OUT + 2 * lane],     accx);
        atomicAdd(&out[cur * F_OUT + 2 * lane + 1], accy);
        accx = 0.f; accy = 0.f;
        cur = r;
      }
      const float2 p = pre2[cN * 32 + lane];   // one b64 gather per lane
      accx = fmaf(v, p.x, accx);
      accy = fmaf(v, p.y, accy);
    }
  }
  atomicAdd(&out[cur * F_OUT + 2 * lane],     accx);
  atomicAdd(&out[cur * F_OUT + 2 * lane + 1], accy);
}

// ---------------------------------------------------------------------------
// Helpers: zero-init of d_out (harness poisons it) and in-place ReLU.
// ---------------------------------------------------------------------------
__global__ __launch_bounds__(256) void gc_zero(float4* __restrict__ p, int n4)
{
  const int i = (int)(blockIdx.x * blockDim.x + threadIdx.x);
  if (i < n4) p[i] = make_float4(0.f, 0.f, 0.f, 0.f);
}

__global__ __launch_bounds__(256) void gc_relu(float4* __restrict__ p, int n4)
{
  const int i = (int)(blockIdx.x * blockDim.x + threadIdx.x);
  if (i < n4) {
    float4 v = p[i];
    v.x = fmaxf(v.x, 0.f); v.y = fmaxf(v.y, 0.f);
    v.z = fmaxf(v.z, 0.f); v.w = fmaxf(v.w, 0.f);
    p[i] = v;
  }
}

extern "C" void kernel_launch(void* const* d_in, const int* in_sizes, int n_in,
                              void* d_out, int out_size, void* d_ws, size_t ws_size,
                              hipStream_t stream)
{
  const float* x    = (const float*)d_in[0];   // [N_NODES, F_IN]
  const float* w    = (const float*)d_in[1];   // [F_IN, F_OUT]
  const int*   erow = (const int*)  d_in[2];   // [N_EDGES] sorted
  const int*   ecol = (const int*)  d_in[3];   // [N_EDGES]
  const float* eval = (const float*)d_in[4];   // [N_EDGES]
  float* out = (float*)d_out;                  // [N_NODES, F_OUT]
  float* pre = (float*)d_ws;                   // scratch: [N_NODES, F_OUT]

  const int n_out = N_NODES * F_OUT;           // 3,200,000 (divisible by 4)
  const int n4    = n_out / 4;

  // 1) zero accumulator (d_out)
  gc_zero<<<(n4 + 255) / 256, 256, 0, stream>>>((float4*)out, n4);

  // 2) dense GEMM -> pre_sup in scratch (12500 tiles, 8 waves/block)
  const int tiles  = (N_NODES / 16) * (F_OUT / 16);
  const int blocks = (tiles + 7) / 8;
  gc_gemm_wmma<<<blocks, 256, 0, stream>>>(x, w, pre);

  // 3) sparse aggregation with segment-run accumulation
  const int waves  = N_EDGES / EDGES_PER_WAVE;   // 6250
  const int ablk   = (waves + 7) / 8;
  gc_aggregate<<<ablk, 256, 0, stream>>>(pre, erow, ecol, eval, out);

  // 4) in-place ReLU
  gc_relu<<<(n4 + 255) / 256, 256, 0, stream>>>((float4*)out, n4);
}